// PartDecoder_35682588295430
// MI455X (gfx1250) — compile-verified
//
#include <hip/hip_runtime.h>

// ---------------------------------------------------------------------------
// Problem constants (match setup_inputs in the reference)
// ---------------------------------------------------------------------------
constexpr int Bn   = 16;    // batch
constexpr int Nn   = 4096;  // points
constexpr int Fn   = 128;   // feat dim
constexpr int Hn   = 1024;  // hidden
constexpr int Mout = 128;   // recon_M

constexpr int MT     = 64;        // rows (b,n pairs) per workgroup in the GEMM
constexpr int NTHR   = 512;       // 16 waves: 2 M-groups x 8 N-groups
constexpr int XPITCH = Hn + 4;    // LDS row pitch (floats): bank = (4m+k)%64 -> conflict free

typedef __attribute__((ext_vector_type(2))) float v2f;
typedef __attribute__((ext_vector_type(8))) float v8f;

__device__ __forceinline__ float leaky(float v) { return v >= 0.f ? v : 0.01f * v; }

// ---------------------------------------------------------------------------
// Kernel 1: hf'[b][h] = b1[h] + sum_f feat[b][f] * W1[f][h]    ([16 x 1024])
// ---------------------------------------------------------------------------
__global__ __launch_bounds__(256)
void hf_kernel(const float* __restrict__ feat, const float* __restrict__ W1,
               const float* __restrict__ b1, float* __restrict__ hf)
{
    int idx = blockIdx.x * 256 + threadIdx.x;     // B*H = 16384 threads
    int b = idx >> 10;
    int h = idx & (Hn - 1);
    float s = b1[h];
    const float* fr = feat + b * Fn;
    #pragma unroll 8
    for (int f = 0; f < Fn; ++f) s += fr[f] * W1[f * Hn + h];
    hf[idx] = s;
}

// ---------------------------------------------------------------------------
// Kernel 1b: repack W2 [H][H] -> W2p [H/4][H][4]  (K-interleaved so a WMMA
// lane's two B dwords are adjacent -> single global_load_b64 in the hot loop)
//   W2p[((k4*H)+col)*4 + r] = W2[(4*k4+r)*H + col]
// ---------------------------------------------------------------------------
__global__ __launch_bounds__(256)
void pack_w2_kernel(const float* __restrict__ W2, float* __restrict__ W2p)
{
    int i = blockIdx.x * 256 + threadIdx.x;       // H*H = 1M elements
    int k4  = i >> 12;                            // / (H*4/1024) : i / 4096
    int rem = i & 4095;
    int col = rem >> 2;
    int r   = rem & 3;
    W2p[i] = W2[(size_t)(4 * k4 + r) * Hn + col];
}

// ---------------------------------------------------------------------------
// Kernel 2: fused  X = leaky(hf' + pc@W1[128:131])  ->  y = leaky(X@W2 + b2)
//           -> net3 = y@W3 + b3.   WMMA f32 16x16x4, output-stationary.
// One WG = 64 rows x all 1024 cols. 16 waves: wave w -> M-group (w>>3) of 32
// rows (2 subtiles), N-group (w&7) of 128 cols (8 tiles of 16).
// ---------------------------------------------------------------------------
__global__ __launch_bounds__(NTHR)
void fused_mlp_kernel(const float* __restrict__ hf,   // [B][H] (b1 folded in)
                      const float* __restrict__ pc,   // [N][3]
                      const float* __restrict__ W1,   // [F+3][H] (rows F..F+2 used)
                      const float* __restrict__ W2p,  // [H/4][H][4] packed
                      const float* __restrict__ b2,   // [H]
                      const float* __restrict__ W3,   // [H][3]
                      const float* __restrict__ b3,   // [3]
                      float* __restrict__ net3)       // [B*N][3]
{
    extern __shared__ float smem[];
    float* X   = smem;                      // MT * XPITCH floats (~257 KB)
    float* pcl = X + MT * XPITCH;           // MT*3
    float* o3  = pcl + MT * 3;              // MT*3 accumulation tile

    const int tid  = threadIdx.x;
    const int row0 = blockIdx.x * MT;       // 4096 % 64 == 0 -> single b per tile
    const int b    = row0 / Nn;
    const int n0   = row0 % Nn;
    const float* hfb = hf + b * Hn;
    const float* w1x = W1 + (Fn + 0) * Hn;
    const float* w1y = W1 + (Fn + 1) * Hn;
    const float* w1z = W1 + (Fn + 2) * Hn;

    if (tid < MT * 3) {
        pcl[tid] = pc[(n0 + tid / 3) * 3 + (tid % 3)];
        o3[tid]  = 0.f;
    }
    __syncthreads();

    // Build X tile in LDS: X[m][k] = leaky(hf'[b][k] + pc[n].W1rows)
    for (int idx = tid; idx < MT * Hn; idx += NTHR) {
        int m = idx >> 10;
        int k = idx & (Hn - 1);
        float v = hfb[k] + pcl[m*3+0]*w1x[k] + pcl[m*3+1]*w1y[k] + pcl[m*3+2]*w1z[k];
        X[m * XPITCH + k] = leaky(v);
    }
    __syncthreads();

    const int wave    = tid >> 5;
    const int lane    = tid & 31;
    const int lhalf   = lane & 15;
    const int khalf   = (lane >> 4) * 2;     // 0 (lanes 0-15) or 2 (lanes 16-31)
    const int wgM     = wave >> 3;           // 0/1 : rows [wgM*32, wgM*32+32)
    const int wgN     = wave & 7;            // cols [wgN*128, wgN*128+128)
    const int mbase   = wgM * 32;
    const int colbase = wgN * 128;

    const v8f vzero = {};
    v8f acc[2][8];
    #pragma unroll
    for (int s = 0; s < 2; ++s)
        #pragma unroll
        for (int t = 0; t < 8; ++t) acc[s][t] = vzero;

    // A-fragment lane bases (ISA 16x4 f32 layout: lanes 0-15 -> K=0,1; 16-31 -> K=2,3)
    const float* a0p = &X[(mbase + lhalf)      * XPITCH + khalf];
    const float* a1p = &X[(mbase + lhalf + 16) * XPITCH + khalf];
    // Packed-B lane base: W2p[((k4*H)+col)*4 + khalf], col = colbase + t*16 + lhalf
    const float* bp  = W2p + (size_t)(colbase + lhalf) * 4 + khalf;

    for (int k4 = 0; k4 < Hn / 4; ++k4) {
        const int k = k4 * 4;
        v2f a0, a1;
        a0.x = a0p[k];     a0.y = a0p[k + 1];      // 8B-aligned pair in LDS
        a1.x = a1p[k];     a1.y = a1p[k + 1];
        const float* bk = bp + (size_t)k4 * (Hn * 4);
        if (k4 + 1 < Hn / 4)
            __builtin_prefetch(bk + Hn * 4, 0, 3); // WGP-scope prefetch of next K-slab
        #pragma unroll
        for (int t = 0; t < 8; ++t) {
            v2f bf = *(const v2f*)(bk + t * 64);   // one global_load_b64 per tile
            acc[0][t] = __builtin_amdgcn_wmma_f32_16x16x4_f32(
                false, a0, false, bf, (short)0, acc[0][t], false, false);
            acc[1][t] = __builtin_amdgcn_wmma_f32_16x16x4_f32(
                false, a1, false, bf, (short)0, acc[1][t], false, false);
        }
    }

    // Epilogue: y = leaky(acc + b2); partial out3 = y @ W3 reduced over columns.
    #pragma unroll
    for (int s = 0; s < 2; ++s) {
        #pragma unroll
        for (int t = 0; t < 8; ++t) {
            int col = colbase + t * 16 + lhalf;
            float bb  = b2[col];
            float w30 = W3[col*3+0], w31 = W3[col*3+1], w32 = W3[col*3+2];
            #pragma unroll
            for (int j = 0; j < 8; ++j) {
                int m = mbase + s * 16 + ((lane < 16) ? j : j + 8);  // C/D layout rows
                float y = leaky(acc[s][t][j] + bb);
                float p0 = y * w30, p1 = y * w31, p2 = y * w32;
                #pragma unroll
                for (int off = 8; off >= 1; off >>= 1) {   // reduce 16 column-lanes
                    p0 += __shfl_xor(p0, off, 32);
                    p1 += __shfl_xor(p1, off, 32);
                    p2 += __shfl_xor(p2, off, 32);
                }
                if (lhalf == 0) {                          // lanes 0 and 16
                    atomicAdd(&o3[m*3+0], p0);             // ds_add_f32
                    atomicAdd(&o3[m*3+1], p1);
                    atomicAdd(&o3[m*3+2], p2);
                }
            }
        }
    }
    __syncthreads();
    if (tid < MT * 3) {
        int m = tid / 3, c = tid % 3;
        net3[(size_t)(row0 + m) * 3 + c] = o3[tid] + b3[c];
    }
}

// ---------------------------------------------------------------------------
// Kernel 3: deterministic FPS (seed index 0), one workgroup per batch.
// Matches jnp.argmax first-max-index tie-breaking (contiguous chunks, strict >,
// lower-index-wins in the tree).
// ---------------------------------------------------------------------------
__global__ __launch_bounds__(256)
void fps_kernel(const float* __restrict__ net3, float* __restrict__ out)
{
    __shared__ float mind[Nn];
    __shared__ float rval[256];
    __shared__ int   ridx[256];
    __shared__ float selp[3];

    const int b = blockIdx.x;
    const int t = threadIdx.x;
    const float* base = net3 + (size_t)b * Nn * 3;
    const int c0 = t * 16;                     // contiguous chunk per thread

    float sx = base[0], sy = base[1], sz = base[2];
    for (int j = 0; j < 16; ++j) {
        int n = c0 + j;
        float dx = base[n*3]-sx, dy = base[n*3+1]-sy, dz = base[n*3+2]-sz;
        mind[n] = dx*dx + dy*dy + dz*dz;
    }
    if (t < 3) out[(size_t)(b * Mout) * 3 + t] = base[t];
    __syncthreads();

    for (int i = 1; i < Mout; ++i) {
        float best = -1.f; int bidx = c0;
        for (int j = 0; j < 16; ++j) {
            int n = c0 + j;
            float v = mind[n];
            if (v > best) { best = v; bidx = n; }
        }
        rval[t] = best; ridx[t] = bidx;
        __syncthreads();
        for (int off = 128; off >= 1; off >>= 1) {
            if (t < off) {
                float v2 = rval[t + off]; int i2 = ridx[t + off];
                if (v2 > rval[t] || (v2 == rval[t] && i2 < ridx[t])) {
                    rval[t] = v2; ridx[t] = i2;
                }
            }
            __syncthreads();
        }
        int sel = ridx[0];
        if (t < 3) {
            float v = base[sel*3 + t];
            selp[t] = v;
            out[(size_t)(b * Mout + i) * 3 + t] = v;
        }
        __syncthreads();
        float qx = selp[0], qy = selp[1], qz = selp[2];
        for (int j = 0; j < 16; ++j) {
            int n = c0 + j;
            float dx = base[n*3]-qx, dy = base[n*3+1]-qy, dz = base[n*3+2]-qz;
            float d = dx*dx + dy*dy + dz*dz;
            if (d < mind[n]) mind[n] = d;
        }
        __syncthreads();
    }
}

// ---------------------------------------------------------------------------
// Launcher
// ---------------------------------------------------------------------------
extern "C" void kernel_launch(void* const* d_in, const int* in_sizes, int n_in,
                              void* d_out, int out_size, void* d_ws, size_t ws_size,
                              hipStream_t stream)
{
    const float* feat = (const float*)d_in[0];   // [16,128]
    const float* pc   = (const float*)d_in[1];   // [4096,3]
    const float* W1   = (const float*)d_in[2];   // [131,1024]
    const float* b1   = (const float*)d_in[3];   // [1024]
    const float* W2   = (const float*)d_in[4];   // [1024,1024]
    const float* b2   = (const float*)d_in[5];   // [1024]
    const float* W3   = (const float*)d_in[6];   // [1024,3]
    const float* b3   = (const float*)d_in[7];   // [3]
    (void)in_sizes; (void)n_in; (void)out_size; (void)ws_size;

    float* out  = (float*)d_out;                 // [16,128,3]
    float* hf   = (float*)d_ws;                  // [16,1024]
    float* net3 = hf + Bn * Hn;                  // [16*4096,3]
    float* W2p  = net3 + (size_t)Bn * Nn * 3;    // [H/4][H][4]  (total ws ~4.9 MB)

    hf_kernel<<<(Bn * Hn) / 256, 256, 0, stream>>>(feat, W1, b1, hf);
    pack_w2_kernel<<<(Hn * Hn) / 256, 256, 0, stream>>>(W2, W2p);

    size_t lds = (size_t)(MT * XPITCH + MT * 3 + MT * 3) * sizeof(float);
    fused_mlp_kernel<<<(Bn * Nn) / MT, NTHR, lds, stream>>>(
        hf, pc, W1, W2p, b2, W3, b3, net3);

    fps_kernel<<<Bn, 256, 0, stream>>>(net3, out);
}